// XMoEThresholdRouter_30537217474764
// MI455X (gfx1250) — compile-verified
//
#include <hip/hip_runtime.h>
#include <hip/hip_bf16.h>
#include <stdint.h>

typedef __attribute__((ext_vector_type(2))) float v2f;
typedef __attribute__((ext_vector_type(8))) float v8f;

#define HDIM 1024
#define NEXP 64

// Map float -> uint32 whose unsigned order == float order (handles negatives).
__device__ __forceinline__ unsigned fkey(float f) {
  unsigned u = __float_as_uint(f);
  return (u & 0x80000000u) ? ~u : (u | 0x80000000u);
}

// ---------------------------------------------------------------------------
// Kernel 0: zero the top-1 statistics accumulators (graph-replay safe).
// ---------------------------------------------------------------------------
__global__ void moe_init(float* __restrict__ cntF, float* __restrict__ sumP) {
  int t = threadIdx.x;
  if (t < NEXP) { cntF[t] = 0.0f; sumP[t] = 0.0f; }
}

// ---------------------------------------------------------------------------
// Kernel 1: fused gating GEMM (fp32 WMMA 16x16x4) + per-token routing.
//   block = 256 threads = 8 wave32, each wave: 16 tokens x 64 experts.
// ---------------------------------------------------------------------------
__global__ __launch_bounds__(256)
void moe_gemm_route(const float* __restrict__ x, const float* __restrict__ gw,
                    unsigned* __restrict__ keys, float* __restrict__ cntF,
                    float* __restrict__ sumP, float* __restrict__ outScores,
                    float* __restrict__ outTop1) {
  __shared__ float lds[128 * 65];  // 128 tokens x 64 logits, padded (bank-safe)

  const int tid  = threadIdx.x;
  const int wid  = tid >> 5;
  const int lane = tid & 31;
  const int half = lane >> 4;   // K-half selector (ISA 7.12.2 fp32 layouts)
  const int lm   = lane & 15;   // M for A-frag, N for B-frag
  const int blockTok  = blockIdx.x * 128;
  const int tokenBase = blockTok + wid * 16;

  // A fragment source: x row (token) lm, starting at K-offset 2*half.
  const float* xrow = x + (size_t)(tokenBase + lm) * HDIM + 2 * half;
  // B fragment sources: gate_w rows (expert) g*16+lm, B(k,n)=gw[n][k].
  const float* w0 = gw + (size_t)(0 * 16 + lm) * HDIM + 2 * half;
  const float* w1 = gw + (size_t)(1 * 16 + lm) * HDIM + 2 * half;
  const float* w2 = gw + (size_t)(2 * 16 + lm) * HDIM + 2 * half;
  const float* w3 = gw + (size_t)(3 * 16 + lm) * HDIM + 2 * half;

  v8f acc0 = {}, acc1 = {}, acc2 = {}, acc3 = {};
#pragma unroll 4
  for (int k = 0; k < HDIM; k += 4) {
    v2f a  = *(const v2f*)(xrow + k);
    v2f b0 = *(const v2f*)(w0 + k);
    v2f b1 = *(const v2f*)(w1 + k);
    v2f b2 = *(const v2f*)(w2 + k);
    v2f b3 = *(const v2f*)(w3 + k);
    acc0 = __builtin_amdgcn_wmma_f32_16x16x4_f32(false, a, false, b0, (short)0, acc0, false, false);
    acc1 = __builtin_amdgcn_wmma_f32_16x16x4_f32(false, a, false, b1, (short)0, acc1, false, false);
    acc2 = __builtin_amdgcn_wmma_f32_16x16x4_f32(false, a, false, b2, (short)0, acc2, false, false);
    acc3 = __builtin_amdgcn_wmma_f32_16x16x4_f32(false, a, false, b3, (short)0, acc3, false, false);
  }

  // Dump D tiles: VGPR j -> (M = j + 8*half, N = lm), experts by group.
#pragma unroll
  for (int j = 0; j < 8; j++) {
    int m = wid * 16 + j + 8 * half;
    lds[m * 65 +  0 + lm] = acc0[j];
    lds[m * 65 + 16 + lm] = acc1[j];
    lds[m * 65 + 32 + lm] = acc2[j];
    lds[m * 65 + 48 + lm] = acc3[j];
  }
  __syncthreads();

  // -------- per-token routing: one thread per token (threads 0..127) -------
  if (tid < 128) {
    const int tt = tid;
    const size_t token = (size_t)blockTok + tt;

    float mx = -3.4e38f;
#pragma unroll
    for (int e = 0; e < NEXP; e++) mx = fmaxf(mx, lds[tt * 65 + e]);
    float s = 0.0f;
#pragma unroll
    for (int e = 0; e < NEXP; e++) {
      float v = __expf(lds[tt * 65 + e] - mx);
      lds[tt * 65 + e] = v;
      s += v;
    }
    const float inv = 1.0f / s;

    // Pack keys: high 32 = prob bits (prob >= 0 so uint-ordered),
    // low 6 = (63 - e) so ties break toward smaller e in descending order.
    unsigned long long kv[NEXP];
#pragma unroll
    for (int e = 0; e < NEXP; e++) {
      float p = lds[tt * 65 + e] * inv;
      outScores[token * NEXP + e] = p;  // full probs; kernel2 zeroes non-kept
      kv[e] = ((unsigned long long)__float_as_uint(p) << 32) |
              (unsigned long long)(63 - e);
    }

    // Bitonic sort (ascending), fully unrolled -> static register indexing.
#pragma unroll
    for (int kk = 2; kk <= NEXP; kk <<= 1) {
#pragma unroll
      for (int j = kk >> 1; j > 0; j >>= 1) {
#pragma unroll
        for (int i = 0; i < NEXP; i++) {
          int l = i ^ j;
          if (l > i) {
            bool up = ((i & kk) == 0);
            unsigned long long va = kv[i], vb = kv[l];
            bool sw = up ? (va > vb) : (va < vb);
            if (sw) { kv[i] = vb; kv[l] = va; }
          }
        }
      }
    }

    // Walk descending order: cumsum threshold, R = p - rank, emit R keys.
    float csum = 0.0f;
    int   top1e = 0;
    float top1p = 0.0f;
#pragma unroll
    for (int pos = 0; pos < NEXP; pos++) {
      unsigned long long key = kv[NEXP - 1 - pos];
      float p = __uint_as_float((unsigned)(key >> 32));
      int   e = 63 - (int)(key & 63ull);
      float excl = csum;
      csum += p;
      bool assigned = (excl < 0.9f);  // pos <= k_idx  <=>  exclusive csum < T
      float R = assigned ? (p - (float)(pos + 1)) : -1.0e9f;
      keys[token * NEXP + e] = fkey(R);
      if (pos == 0) { top1e = e; top1p = p; }
    }

    outTop1[token] = (float)top1e;
    atomicAdd(&cntF[top1e], 1.0f);
    atomicAdd(&sumP[top1e], top1p);
  }
}

// ---------------------------------------------------------------------------
// Kernel 2: per-expert top-C selection via 4-pass radix select + stable ties.
//   one 1024-thread block per expert column; thread t owns tokens [t*TPT, ...)
// ---------------------------------------------------------------------------
__global__ __launch_bounds__(1024)
void moe_select(const unsigned* __restrict__ keys, float* __restrict__ outMask,
                float* __restrict__ outScores, int N, int C) {
  const int e = blockIdx.x;
  const int t = threadIdx.x;
  const int TPT = N / 1024;  // 32 tokens per thread, contiguous (token order)

  __shared__ unsigned hist[256];
  __shared__ unsigned sh_prefix;
  __shared__ unsigned sh_crem;
  __shared__ unsigned sc[1024];

  unsigned prefix = 0u, pmask = 0u, crem = (unsigned)C;
  for (int pass = 0; pass < 4; pass++) {
    const int shift = 24 - 8 * pass;
    if (t < 256) hist[t] = 0u;
    __syncthreads();
    for (int i = 0; i < TPT; i++) {
      int tok = t * TPT + i;
      unsigned key = keys[(size_t)tok * NEXP + e];
      if ((key & pmask) == prefix)
        atomicAdd(&hist[(key >> shift) & 255u], 1u);
    }
    __syncthreads();
    if (t == 0) {
      unsigned cum = 0u;
      int b = 255;
      for (; b >= 0; b--) {
        unsigned h = hist[b];
        if (cum + h >= crem) break;  // b is the threshold bin
        cum += h;
      }
      sh_prefix = prefix | ((unsigned)b << shift);
      sh_crem   = crem - cum;  // still-needed count inside threshold bin
    }
    __syncthreads();
    prefix = sh_prefix;
    crem   = sh_crem;
    pmask |= (0xFFu << shift);
    __syncthreads();
  }
  const unsigned thr   = prefix;  // exact 512th-largest key
  const unsigned slots = crem;    // #entries equal to thr to keep (token order)

  // Count equals per thread, block-wide exclusive scan for stable tie slots.
  unsigned eq = 0u;
  for (int i = 0; i < TPT; i++) {
    unsigned key = keys[(size_t)(t * TPT + i) * NEXP + e];
    eq += (key == thr) ? 1u : 0u;
  }
  sc[t] = eq;
  __syncthreads();
  for (int off = 1; off < 1024; off <<= 1) {
    unsigned v = (t >= off) ? sc[t - off] : 0u;
    __syncthreads();
    sc[t] += v;
    __syncthreads();
  }
  unsigned ord = sc[t] - eq;  // exclusive prefix = ordinal of first local equal

  const unsigned KUN = fkey(-1.0e9f);  // unassigned sentinel key
  for (int i = 0; i < TPT; i++) {
    int tok = t * TPT + i;
    size_t off = (size_t)tok * NEXP + e;
    unsigned key = keys[off];
    bool gt  = key > thr;
    bool eqk = (key == thr);
    bool kept = gt || (eqk && ord < slots);
    if (eqk) ord++;
    bool fin = kept && (key != KUN);  // AND with assign mask
    outMask[off] = fin ? 1.0f : 0.0f;
    if (!fin) outScores[off] = 0.0f;  // probs pre-written by kernel 1
  }
}

// ---------------------------------------------------------------------------
// Kernel 3: aux loss = E * sum_e (cnt/N)*(sumP/N) * alpha
// ---------------------------------------------------------------------------
__global__ void moe_aux(const float* __restrict__ cntF,
                        const float* __restrict__ sumP,
                        float* __restrict__ outAux, int N) {
  if (threadIdx.x == 0 && blockIdx.x == 0) {
    float s = 0.0f;
    for (int e = 0; e < NEXP; e++) s += cntF[e] * sumP[e];
    float n = (float)N;
    *outAux = (float)NEXP * (s / (n * n)) * 0.01f;
  }
}

extern "C" void kernel_launch(void* const* d_in, const int* in_sizes, int n_in,
                              void* d_out, int out_size, void* d_ws, size_t ws_size,
                              hipStream_t stream) {
  (void)n_in; (void)out_size; (void)ws_size;
  const float* x  = (const float*)d_in[0];   // (B*T, H) fp32
  const float* gw = (const float*)d_in[1];   // (E, H)  fp32

  const int N = in_sizes[0] / HDIM;          // B*T = 32768
  int C = (N + NEXP - 1) / NEXP;             // ceil(N/E) * CF(=1.0)
  if (C > N) C = N;

  float* out       = (float*)d_out;
  float* outMask   = out;                              // N*E floats
  float* outScores = out + (size_t)N * NEXP;           // N*E floats
  float* outAux    = out + (size_t)2 * N * NEXP;       // 1 float
  float* outTop1   = outAux + 1;                       // N floats

  unsigned* keys = (unsigned*)d_ws;                                    // N*E u32
  float* cntF = (float*)((char*)d_ws + (size_t)N * NEXP * sizeof(unsigned));
  float* sumP = cntF + NEXP;

  hipLaunchKernelGGL(moe_init, dim3(1), dim3(64), 0, stream, cntF, sumP);
  hipLaunchKernelGGL(moe_gemm_route, dim3(N / 128), dim3(256), 0, stream,
                     x, gw, keys, cntF, sumP, outScores, outTop1);
  hipLaunchKernelGGL(moe_select, dim3(NEXP), dim3(1024), 0, stream,
                     keys, outMask, outScores, N, C);
  hipLaunchKernelGGL(moe_aux, dim3(1), dim3(1), 0, stream, cntF, sumP, outAux, N);
}